// TopDownLSTMEncoder_24618752541150
// MI455X (gfx1250) — compile-verified
//
#include <hip/hip_runtime.h>
#include <cstdint>
#include <cstddef>

#define B_  16
#define N_  512
#define I_  512
#define H_  1024

typedef __attribute__((ext_vector_type(2))) float v2f;
typedef __attribute__((ext_vector_type(8))) float v8f;

// ---------------------------------------------------------------------------
// CDNA5 fp32 WMMA: D(16x16,f32) = A(16x4,f32) * B(4x16,f32) + C
//   A: lane = (k>=2)*16 + m, vgpr = k&1
//   B: vgpr v, K = v + (lane>=16)*2, n = lane&15
//   C/D: vgpr r, m = r + 8*(lane>>4), n = lane&15
// ---------------------------------------------------------------------------
__device__ __forceinline__ v8f wmma_k4(v2f a, v2f b, v8f c) {
#if defined(__gfx1250__) && __has_builtin(__builtin_amdgcn_wmma_f32_16x16x4_f32)
  return __builtin_amdgcn_wmma_f32_16x16x4_f32(
      false, a, false, b, (short)0, c, false, false);
#else
  v8f d = c;
  d[0] += a.x * b.x + a.y * b.y;   // host pass only
  return d;
#endif
}

__device__ __forceinline__ float sigmoidf_(float x) {
  return 1.0f / (1.0f + __expf(-x));
}

// ---------------------------------------------------------------------------
// Kernel 1: g0[gate] = X[:,0,:] @ W_gate + b_gate        (3 x 16x1024)
// ---------------------------------------------------------------------------
__global__ void k_g0_gemm(const float* __restrict__ X,
                          const float* __restrict__ Wf, const float* __restrict__ Wo,
                          const float* __restrict__ Wz,
                          const float* __restrict__ bf, const float* __restrict__ bo,
                          const float* __restrict__ bz,
                          float* __restrict__ g0) {
  const int wg = blockIdx.x * (blockDim.x >> 5) + (threadIdx.x >> 5);
  if (wg >= 3 * (H_ / 16)) return;
  const int gate  = wg / (H_ / 16);
  const int ntile = wg % (H_ / 16);
  const int lane  = threadIdx.x & 31;
  const int l15   = lane & 15;
  const int h2    = (lane >> 4) * 2;
  const int col   = ntile * 16 + l15;

  const float* W    = (gate == 0) ? Wf : (gate == 1) ? Wo : Wz;
  const float* bias = (gate == 0) ? bf : (gate == 1) ? bo : bz;
  const float* arow = X + (size_t)l15 * (N_ * I_);   // batch l15, node 0

  v8f acc0 = {}, acc1 = {};
  for (int k0 = 0; k0 < I_; k0 += 8) {
    v2f a0 = *(const v2f*)(arow + k0 + h2);
    v2f a1 = *(const v2f*)(arow + k0 + 4 + h2);
    v2f b0, b1;
    b0.x = W[(size_t)(k0 + h2)     * H_ + col];
    b0.y = W[(size_t)(k0 + h2 + 1) * H_ + col];
    b1.x = W[(size_t)(k0 + 4 + h2)     * H_ + col];
    b1.y = W[(size_t)(k0 + 4 + h2 + 1) * H_ + col];
    acc0 = wmma_k4(a0, b0, acc0);
    acc1 = wmma_k4(a1, b1, acc1);
  }
  v8f acc = acc0 + acc1;
  const float bv = bias[col];
  float* out = g0 + (size_t)gate * (B_ * H_);
#pragma unroll
  for (int r = 0; r < 8; ++r) {
    const int mr = r + 8 * (lane >> 4);
    out[(size_t)mr * H_ + col] = acc[r] + bv;
  }
}

// ---------------------------------------------------------------------------
// Kernel 2: V_gate = T_gateᵀ @ T_gate, stored K-pair interleaved:
//   Vpk[gate][(k>>1)*2H + n*2 + (k&1)] = V[k][n]
// so the scan fetches {V[k][n], V[k+1][n]} with a single coalesced b64.
// ---------------------------------------------------------------------------
__global__ void k_vtv(const float* __restrict__ Tf, const float* __restrict__ To,
                      const float* __restrict__ Tz, float* __restrict__ Vpk) {
  const int tilesPer = (H_ / 16) * (H_ / 16);   // 4096
  const int wg = blockIdx.x * (blockDim.x >> 5) + (threadIdx.x >> 5);
  if (wg >= 3 * tilesPer) return;
  const int gate = wg / tilesPer;
  const int rem  = wg % tilesPer;
  const int i0   = (rem / (H_ / 16)) * 16;
  const int j0   = (rem % (H_ / 16)) * 16;
  const int lane = threadIdx.x & 31;
  const int l15  = lane & 15;
  const int h2   = (lane >> 4) * 2;

  const float* T  = (gate == 0) ? Tf : (gate == 1) ? To : Tz;
  float*       Vg = Vpk + (size_t)gate * H_ * H_;

  v8f acc0 = {}, acc1 = {};
  for (int k0 = 0; k0 < H_; k0 += 8) {
    v2f a0, b0, a1, b1;
    a0.x = T[(size_t)(k0 + h2)     * H_ + i0 + l15];
    a0.y = T[(size_t)(k0 + h2 + 1) * H_ + i0 + l15];
    b0.x = T[(size_t)(k0 + h2)     * H_ + j0 + l15];
    b0.y = T[(size_t)(k0 + h2 + 1) * H_ + j0 + l15];
    a1.x = T[(size_t)(k0 + 4 + h2)     * H_ + i0 + l15];
    a1.y = T[(size_t)(k0 + 4 + h2 + 1) * H_ + i0 + l15];
    b1.x = T[(size_t)(k0 + 4 + h2)     * H_ + j0 + l15];
    b1.y = T[(size_t)(k0 + 4 + h2 + 1) * H_ + j0 + l15];
    acc0 = wmma_k4(a0, b0, acc0);
    acc1 = wmma_k4(a1, b1, acc1);
  }
  v8f acc = acc0 + acc1;
#pragma unroll
  for (int r = 0; r < 8; ++r) {
    const int mr  = r + 8 * (lane >> 4);
    const int row = i0 + mr;                 // K index of the scan
    const int cj  = j0 + l15;                // N index of the scan
    Vg[(size_t)(row >> 1) * (2 * H_) + cj * 2 + (row & 1)] = acc[r];
  }
}

// ---------------------------------------------------------------------------
// Kernel 3: root node (node 0) elementwise init of tree_h / tree_c.
// ---------------------------------------------------------------------------
__global__ void k_root(const float* __restrict__ g0,
                       float* __restrict__ tree_h, float* __restrict__ tree_c) {
  const int idx = blockIdx.x * blockDim.x + threadIdx.x;
  if (idx >= B_ * H_) return;
  const int b = idx / H_, col = idx % H_;
  const float f = sigmoidf_(g0[(size_t)b * H_ + col]);
  const float o = sigmoidf_(g0[(size_t)(B_ * H_) + b * H_ + col]);
  const float z = tanhf(g0[(size_t)(2 * B_ * H_) + b * H_ + col]);
  const float c = z * (1.0f - f);
  const float h = o * tanhf(c);
  tree_h[((size_t)b * N_) * H_ + col] = h;
  tree_c[((size_t)b * N_) * H_ + col] = c;
}

// ---------------------------------------------------------------------------
// Kernel 4: persistent scan with producer/consumer row-ready flags.
// 64 blocks x 96 threads: block = one 16-column N-tile, one wave per gate.
// No global lockstep barrier: row_cnt[i] counts the 64 tile-chunks of node
// row i that have been published; consumers spin only on the parent rows
// they actually gather, so blocks pipeline across steps.
// ---------------------------------------------------------------------------
#define SCAN_BLOCKS  64
#define SCAN_THREADS 96

__global__ void __launch_bounds__(SCAN_THREADS, 1)
k_scan(const int* __restrict__ conn, const float* __restrict__ Vpk,
       const float* __restrict__ g0, float* __restrict__ tree_h,
       float* __restrict__ tree_c, unsigned int* __restrict__ row_cnt) {
  const int wave  = threadIdx.x >> 5;        // 0..2  (gate f/o/z)
  const int lane  = threadIdx.x & 31;
  const int gate  = wave;
  const int ntile = blockIdx.x;              // 0..63
  const int l15   = lane & 15;
  const int hi    = lane >> 4;
  const int h2    = hi * 2;
  const int col   = ntile * 16 + l15;

  const float* Vg = Vpk + (size_t)gate * H_ * H_;
  const float* f0 = g0;
  const float* o0 = g0 + (size_t)B_ * H_;
  const float* z0 = g0 + (size_t)2 * B_ * H_;

  __shared__ float lds[2][2][256];  // [step parity][gate-1][r*32+lane]

  for (int i = 1; i < N_; ++i) {
    // --- wait until this lane's parent row is fully published -------------
    const int par = conn[l15 * N_ + i];
    if (par > 0) {
      while (__hip_atomic_load(&row_cnt[par], __ATOMIC_ACQUIRE,
                               __HIP_MEMORY_SCOPE_AGENT) < (unsigned)SCAN_BLOCKS) {
        __builtin_amdgcn_s_sleep(1);
      }
    }
    __threadfence();   // acquire: drop stale L0 lines before the gather

    const float* arow = tree_h + ((size_t)l15 * N_ + par) * H_;

    // --- 16x16 tile of  ph @ V_gate : 4 independent WMMA chains -----------
    v8f a0 = {}, a1 = {}, a2 = {}, a3 = {};
    for (int k0 = 0; k0 < H_; k0 += 16) {
      v2f x0 = *(const v2f*)(arow + k0 + h2);
      v2f x1 = *(const v2f*)(arow + k0 + 4 + h2);
      v2f x2 = *(const v2f*)(arow + k0 + 8 + h2);
      v2f x3 = *(const v2f*)(arow + k0 + 12 + h2);
      v2f y0 = *(const v2f*)(Vg + (size_t)((k0 + h2) >> 1)      * (2 * H_) + col * 2);
      v2f y1 = *(const v2f*)(Vg + (size_t)((k0 + 4 + h2) >> 1)  * (2 * H_) + col * 2);
      v2f y2 = *(const v2f*)(Vg + (size_t)((k0 + 8 + h2) >> 1)  * (2 * H_) + col * 2);
      v2f y3 = *(const v2f*)(Vg + (size_t)((k0 + 12 + h2) >> 1) * (2 * H_) + col * 2);
      a0 = wmma_k4(x0, y0, a0);
      a1 = wmma_k4(x1, y1, a1);
      a2 = wmma_k4(x2, y2, a2);
      a3 = wmma_k4(x3, y3, a3);
    }
    v8f acc = (a0 + a1) + (a2 + a3);

    if (gate != 0) {
#pragma unroll
      for (int r = 0; r < 8; ++r) lds[i & 1][gate - 1][r * 32 + lane] = acc[r];
    }
    __syncthreads();

    if (gate == 0) {
#pragma unroll
      for (int r = 0; r < 8; ++r) {
        const int mr = r + 8 * hi;  // batch index
        const float xf = acc[r]                       + f0[(size_t)mr * H_ + col];
        const float xo = lds[i & 1][0][r * 32 + lane] + o0[(size_t)mr * H_ + col];
        const float xz = lds[i & 1][1][r * 32 + lane] + z0[(size_t)mr * H_ + col];
        const float fg = sigmoidf_(xf);
        const float og = sigmoidf_(xo);
        const float zg = tanhf(xz);
        const int   pr = conn[mr * N_ + i];
        const float pc = tree_c[((size_t)mr * N_ + pr) * H_ + col];
        const float c  = pc * fg + zg * (1.0f - fg);
        const float h  = og * tanhf(c);
        tree_c[((size_t)mr * N_ + i) * H_ + col] = c;
        tree_h[((size_t)mr * N_ + i) * H_ + col] = h;
      }
      __threadfence();   // release: flush h/c chunk to device scope
      if (lane == 0) {
        __hip_atomic_fetch_add(&row_cnt[i], 1u, __ATOMIC_RELEASE,
                               __HIP_MEMORY_SCOPE_AGENT);
      }
    }
    // LDS parity double-buffering: one __syncthreads per step is enough —
    // waves stay within one step of each other, so buffer (i&1) is not
    // rewritten until its readers passed the next step's barrier.
  }
}

// ---------------------------------------------------------------------------
extern "C" void kernel_launch(void* const* d_in, const int* in_sizes, int n_in,
                              void* d_out, int out_size, void* d_ws, size_t ws_size,
                              hipStream_t stream) {
  const float* X    = (const float*)d_in[0];
  const int*   conn = (const int*)d_in[1];
  // d_in[2] node_mask: all ones, unused by the reference output
  const float* Wf = (const float*)d_in[3];
  const float* bf = (const float*)d_in[4];
  const float* Wo = (const float*)d_in[5];
  const float* bo = (const float*)d_in[6];
  const float* Wz = (const float*)d_in[7];
  const float* bz = (const float*)d_in[8];
  const float* Tf = (const float*)d_in[9];
  const float* To = (const float*)d_in[10];
  const float* Tz = (const float*)d_in[11];

  float* tree_h = (float*)d_out;

  char*  ws  = (char*)d_ws;
  size_t off = 0;
  float* Vpk    = (float*)(ws + off); off += (size_t)3 * H_ * H_ * sizeof(float);
  float* g0     = (float*)(ws + off); off += (size_t)3 * B_ * H_ * sizeof(float);
  float* tree_c = (float*)(ws + off); off += (size_t)B_ * N_ * H_ * sizeof(float);
  off = (off + 255) & ~(size_t)255;
  unsigned int* row_cnt = (unsigned int*)(ws + off);

  hipMemsetAsync(row_cnt, 0, N_ * sizeof(unsigned int), stream);

  {  // g0 = X[:,0,:] @ W + b   : 192 waves
    const int waves  = 3 * (H_ / 16);
    const int blocks = (waves + 7) / 8;
    k_g0_gemm<<<blocks, 256, 0, stream>>>(X, Wf, Wo, Wz, bf, bo, bz, g0);
  }
  {  // V = Tᵀ T (packed) : 12288 waves
    const int waves  = 3 * (H_ / 16) * (H_ / 16);
    const int blocks = (waves + 7) / 8;
    k_vtv<<<blocks, 256, 0, stream>>>(Tf, To, Tz, Vpk);
  }
  k_root<<<(B_ * H_ + 255) / 256, 256, 0, stream>>>(g0, tree_h, tree_c);
  k_scan<<<SCAN_BLOCKS, SCAN_THREADS, 0, stream>>>(conn, Vpk, g0, tree_h,
                                                   tree_c, row_cnt);
}